// CausalLinearAttention_46325517254670
// MI455X (gfx1250) — compile-verified
//
#include <hip/hip_runtime.h>
#include <hip/hip_bf16.h>

// Causal linear attention (elu+1 feature map), chunk-parallel 3-phase scheme.
// MI455X / gfx1250: fp32 WMMA 16x16x4, wave32, 256-thread (8-wave) blocks.

#define N_  4
#define L_  4096
#define H_  8
#define D_  64
#define DV_ 64
#define C_  64            // chunk length
#define T_  (L_ / C_)     // chunks per sequence = 64
#define NH_ (N_ * H_)     // 32
#define EPS_ 1e-6f

typedef float v2f __attribute__((ext_vector_type(2)));
typedef float v8f __attribute__((ext_vector_type(8)));

__device__ __forceinline__ v8f wmma4(v2f a, v2f b, v8f c) {
  // D = A(16x4,f32) * B(4x16,f32) + C(16x16,f32)
  return __builtin_amdgcn_wmma_f32_16x16x4_f32(
      /*neg_a=*/false, a, /*neg_b=*/false, b,
      /*c_mod=*/(short)0, c, /*reuse_a=*/false, /*reuse_b=*/false);
}

// A fragment, row-major A[m][k] = buf[m*ld + k]
__device__ __forceinline__ v2f frag_a_rm(const float* buf, int ld, int mb, int kk, int lane) {
  int m = mb + (lane & 15);
  int k = kk * 4 + ((lane >> 4) << 1);   // lanes 16-31 hold K=2,3
  v2f a; a.x = buf[m * ld + k]; a.y = buf[m * ld + k + 1]; return a;
}
// A fragment, transposed storage: A[m][k] = buf[k*ld + m]
__device__ __forceinline__ v2f frag_a_tr(const float* buf, int ld, int mb, int kk, int lane) {
  int m = mb + (lane & 15);
  int k = kk * 4 + ((lane >> 4) << 1);
  v2f a; a.x = buf[k * ld + m]; a.y = buf[(k + 1) * ld + m]; return a;
}
// B fragment, row-major B[k][n] = buf[k*ld + n]
__device__ __forceinline__ v2f frag_b_rm(const float* buf, int ld, int nb, int kk, int lane) {
  int n = nb + (lane & 15);
  int k = kk * 4 + ((lane >> 4) << 1);
  v2f b; b.x = buf[k * ld + n]; b.y = buf[(k + 1) * ld + n]; return b;
}
// B fragment, transposed storage: B[k][n] = buf[n*ld + k]
__device__ __forceinline__ v2f frag_b_tr(const float* buf, int ld, int nb, int kk, int lane) {
  int n = nb + (lane & 15);
  int k = kk * 4 + ((lane >> 4) << 1);
  v2f b; b.x = buf[n * ld + k]; b.y = buf[n * ld + k + 1]; return b;
}

__device__ __forceinline__ float featmap(float x) {   // elu(x)+1
  return x > 0.0f ? x + 1.0f : __expf(x);
}

// ---------------- Phase 1: per-chunk  KV = K^T V  and ksum = colsum(K) -------
__global__ __launch_bounds__(256) void cla_phase1(
    const float* __restrict__ kin, const float* __restrict__ vin,
    float* __restrict__ kv, float* __restrict__ ksum) {
  __shared__ float Ks[C_ * D_];
  __shared__ float Vs[C_ * DV_];

  const int bt = blockIdx.x;
  const int nh = bt / T_, t = bt % T_;
  const int n = nh / H_, h = nh % H_;
  const int tid = threadIdx.x;
  const size_t rs = (size_t)H_ * D_;  // row stride in [N,L,H,D]
  const float* kb = kin + (((size_t)n * L_ + (size_t)t * C_) * H_ + h) * D_;
  const float* vb = vin + (((size_t)n * L_ + (size_t)t * C_) * H_ + h) * D_;

  for (int idx = tid; idx < C_ * D_; idx += 256) {
    int r = idx >> 6, c = idx & 63;
    Ks[idx] = featmap(kb[(size_t)r * rs + c]);
    Vs[idx] = vb[(size_t)r * rs + c];
  }
  __syncthreads();

  const int wave = tid >> 5, lane = tid & 31;
  float* kvb = kv + (size_t)(nh * T_ + t) * (D_ * DV_);

  // 16 output tiles (4x4 of 16x16) over D x Dv; 2 tiles per wave.
#pragma unroll
  for (int j = 0; j < 2; ++j) {
    const int tt = wave * 2 + j;
    const int mb = (tt >> 2) * 16, nb = (tt & 3) * 16;
    v8f acc = {};
#pragma unroll
    for (int kk = 0; kk < 16; ++kk) {               // k-dim = chunk rows (64)
      v2f a = frag_a_tr(Ks, D_, mb, kk, lane);      // A = K^T : A[d][l] = Ks[l*D+d]
      v2f b = frag_b_rm(Vs, DV_, nb, kk, lane);     // B = V
      acc = wmma4(a, b, acc);
    }
    const int col = nb + (lane & 15);
    const int rb  = mb + ((lane >> 4) << 3);
#pragma unroll
    for (int r = 0; r < 8; ++r)
      kvb[(size_t)(rb + r) * DV_ + col] = acc[r];
  }

  if (tid < D_) {
    float s = 0.0f;
    for (int r = 0; r < C_; ++r) s += Ks[r * D_ + tid];
    ksum[(size_t)(nh * T_ + t) * D_ + tid] = s;
  }
}

// ---------------- Phase 2: in-place exclusive prefix scan over chunks --------
__global__ __launch_bounds__(256) void cla_phase2(
    float* __restrict__ kv, float* __restrict__ ksum) {
  const int nh = blockIdx.x;
  const int tid = threadIdx.x;
  float run[16];
#pragma unroll
  for (int i = 0; i < 16; ++i) run[i] = 0.0f;
  float runk = 0.0f;

  float* base  = kv + (size_t)nh * T_ * (D_ * DV_);
  float* kbase = ksum + (size_t)nh * T_ * D_;

  for (int t = 0; t < T_; ++t) {
    float* p = base + (size_t)t * (D_ * DV_);
#pragma unroll
    for (int i = 0; i < 16; ++i) {
      const int idx = tid + i * 256;
      const float cur = p[idx];
      p[idx] = run[i];          // exclusive prefix
      run[i] += cur;
    }
    if (tid < D_) {
      const float cur = kbase[t * D_ + tid];
      kbase[t * D_ + tid] = runk;
      runk += cur;
    }
  }
}

// ---------------- Phase 3: out = (mask o QK^T) V + Q S_prefix, normalized ----
__global__ __launch_bounds__(256) void cla_phase3(
    const float* __restrict__ qin, const float* __restrict__ kin,
    const float* __restrict__ vin, const float* __restrict__ kv,
    const float* __restrict__ ksum, float* __restrict__ out) {
  __shared__ float Qs[C_ * D_];
  __shared__ float Vs[C_ * DV_];
  __shared__ float Bs[C_ * D_];   // reused: S_prefix -> K -> masked P
  __shared__ float ks[D_];
  __shared__ float inv[C_];

  const int bt = blockIdx.x;
  const int nh = bt / T_, t = bt % T_;
  const int n = nh / H_, h = nh % H_;
  const int tid = threadIdx.x;
  const size_t rs = (size_t)H_ * D_;
  const float* qb = qin + (((size_t)n * L_ + (size_t)t * C_) * H_ + h) * D_;
  const float* kb = kin + (((size_t)n * L_ + (size_t)t * C_) * H_ + h) * D_;
  const float* vb = vin + (((size_t)n * L_ + (size_t)t * C_) * H_ + h) * D_;
  const float* sp = kv + (size_t)(nh * T_ + t) * (D_ * DV_);

  for (int idx = tid; idx < C_ * D_; idx += 256) {
    int r = idx >> 6, c = idx & 63;
    Qs[idx] = featmap(qb[(size_t)r * rs + c]);
    Vs[idx] = vb[(size_t)r * rs + c];
    Bs[idx] = sp[idx];                 // S_prefix (D x Dv)
  }
  if (tid < D_) ks[tid] = ksum[(size_t)(nh * T_ + t) * D_ + tid];
  __syncthreads();

  const int wave = tid >> 5, lane = tid & 31;
  v8f acc[2];

  // ---- acc = Q @ S_prefix (A = Q row-major, B = S row-major) ----
#pragma unroll
  for (int j = 0; j < 2; ++j) {
    const int tt = wave * 2 + j;
    const int mb = (tt >> 2) * 16, nb = (tt & 3) * 16;
    acc[j] = (v8f){};
#pragma unroll
    for (int kk = 0; kk < 16; ++kk) {
      v2f a = frag_a_rm(Qs, D_, mb, kk, lane);
      v2f b = frag_b_rm(Bs, DV_, nb, kk, lane);
      acc[j] = wmma4(a, b, acc[j]);
    }
  }
  __syncthreads();                     // done reading S from Bs

  for (int idx = tid; idx < C_ * D_; idx += 256) {
    int r = idx >> 6, c = idx & 63;
    Bs[idx] = featmap(kb[(size_t)r * rs + c]);   // K (feature-mapped)
  }
  __syncthreads();

  // ---- P = Q @ K^T; skip strictly-upper tiles (wave-uniform) ----
  v8f p[2];
#pragma unroll
  for (int j = 0; j < 2; ++j) {
    const int tt = wave * 2 + j;
    const int ti = tt >> 2, tj = tt & 3;
    p[j] = (v8f){};
    if (tj <= ti) {
#pragma unroll
      for (int kk = 0; kk < 16; ++kk) {          // k-dim = feature dim D
        v2f a = frag_a_rm(Qs, D_, ti * 16, kk, lane);
        v2f b = frag_b_tr(Bs, D_, tj * 16, kk, lane);  // B = K^T : B[d][l2]=Ks[l2*D+d]
        p[j] = wmma4(a, b, p[j]);
      }
    }
  }
  __syncthreads();                     // done reading K from Bs

  // ---- write causal-masked P back into Bs ----
#pragma unroll
  for (int j = 0; j < 2; ++j) {
    const int tt = wave * 2 + j;
    const int ti = tt >> 2, tj = tt & 3;
    const int col = tj * 16 + (lane & 15);
    const int rb  = ti * 16 + ((lane >> 4) << 3);
#pragma unroll
    for (int r = 0; r < 8; ++r) {
      const int row = rb + r;
      Bs[row * C_ + col] = (col <= row) ? p[j][r] : 0.0f;
    }
  }
  __syncthreads();

  // ---- normalizer: z[l] = rowsum(maskP)[l] + Q[l].ksum_prefix + eps ----
  if (tid < C_) {
    float s = 0.0f, dq = 0.0f;
    for (int c = 0; c < C_; ++c) {
      s  += Bs[tid * C_ + c];
      dq += Qs[tid * D_ + c] * ks[c];
    }
    inv[tid] = 1.0f / (s + dq + EPS_);
  }

  // ---- acc += maskP @ V ----
#pragma unroll
  for (int j = 0; j < 2; ++j) {
    const int tt = wave * 2 + j;
    const int mb = (tt >> 2) * 16, nb = (tt & 3) * 16;
#pragma unroll
    for (int kk = 0; kk < 16; ++kk) {            // k-dim = chunk cols (64)
      v2f a = frag_a_rm(Bs, C_, mb, kk, lane);
      v2f b = frag_b_rm(Vs, DV_, nb, kk, lane);
      acc[j] = wmma4(a, b, acc[j]);
    }
  }
  __syncthreads();                     // inv[] visible to all

  float* ob = out + (((size_t)n * L_ + (size_t)t * C_) * H_ + h) * DV_;
#pragma unroll
  for (int j = 0; j < 2; ++j) {
    const int tt = wave * 2 + j;
    const int mb = (tt >> 2) * 16, nb = (tt & 3) * 16;
    const int col = nb + (lane & 15);
    const int rb  = mb + ((lane >> 4) << 3);
#pragma unroll
    for (int r = 0; r < 8; ++r) {
      const int row = rb + r;
      ob[(size_t)row * (H_ * DV_) + col] = acc[j][r] * inv[row];
    }
  }
}

extern "C" void kernel_launch(void* const* d_in, const int* in_sizes, int n_in,
                              void* d_out, int out_size, void* d_ws, size_t ws_size,
                              hipStream_t stream) {
  (void)in_sizes; (void)n_in; (void)out_size; (void)ws_size;
  const float* q = (const float*)d_in[0];
  const float* k = (const float*)d_in[1];
  const float* v = (const float*)d_in[2];
  float* out = (float*)d_out;

  // workspace: per-chunk KV states (exclusive-prefixed in place) + K colsums
  float* kv   = (float*)d_ws;                               // NH*T*D*Dv floats (~33.6 MB)
  float* ksum = kv + (size_t)NH_ * T_ * D_ * DV_;           // NH*T*D floats (~0.5 MB)

  cla_phase1<<<NH_ * T_, 256, 0, stream>>>(k, v, kv, ksum);
  cla_phase2<<<NH_,       256, 0, stream>>>(kv, ksum);
  cla_phase3<<<NH_ * T_, 256, 0, stream>>>(q, k, v, kv, ksum, out);
}